// MonitorRetina_63204738728023
// MI455X (gfx1250) — compile-verified
//
#include <hip/hip_runtime.h>
#include <math.h>

typedef __attribute__((ext_vector_type(2))) float v2f;
typedef __attribute__((ext_vector_type(8))) float v8f;

#define N_SAMPLES 256
#define FEAT      32
#define C_IN      3
#define H_IN      270
#define W_IN      480
#define H_OUT     144
#define W_OUT     256

#define GAMMA     1.7015043497085571f
#define DEG2RAD_FOV 0.6544984694978736f   // 37.5 deg in radians

__device__ __forceinline__ float gelu_gamma(float x) {
    // jax.nn.gelu default (approximate=True, tanh form), then variance gain
    const float k0 = 0.7978845608028654f;   // sqrt(2/pi)
    const float k1 = 0.044715f;
    float t = tanhf(k0 * (x + k1 * x * x * x));
    return 0.5f * x * (1.0f + t) * GAMMA;
}

// ---------------------------------------------------------------------------
// Kernel A: per-sample MLP -> angles -> 3x3 rotation matrices.
// One wave (32 threads) per 16-sample tile; matmuls via V_WMMA_F32_16X16X4_F32.
// ---------------------------------------------------------------------------
__launch_bounds__(32)
__global__ void MonitorRetina_mlp_rmat(const float* __restrict__ eye,
                                       const float* __restrict__ W0, const float* __restrict__ b0,
                                       const float* __restrict__ W1, const float* __restrict__ b1,
                                       const float* __restrict__ Wp, const float* __restrict__ bp,
                                       float* __restrict__ Rout) {
    __shared__ float act[16][FEAT];
    __shared__ float angb[16][3];

    const int lane = threadIdx.x;        // 0..31
    const int half = lane >> 4;          // 0: K/M cols 0-1 half, 1: 2-3 half
    const int l15  = lane & 15;
    const int m0   = blockIdx.x * 16;    // sample tile base

    // ---- layer 0: [16,2] @ [2,32], K padded 2->4 ----------------------------
    v2f a;
    a.x = (half == 0) ? eye[(m0 + l15) * 2 + 0] : 0.0f;
    a.y = (half == 0) ? eye[(m0 + l15) * 2 + 1] : 0.0f;

    v8f acc0[2];
    #pragma unroll
    for (int j = 0; j < 2; ++j) {
        int n = j * 16 + l15;
        v2f b;
        b.x = (half == 0) ? W0[0 * FEAT + n] : 0.0f;
        b.y = (half == 0) ? W0[1 * FEAT + n] : 0.0f;
        v8f c = {};
        acc0[j] = __builtin_amdgcn_wmma_f32_16x16x4_f32(false, a, false, b,
                                                        (short)0, c, false, false);
    }
    #pragma unroll
    for (int j = 0; j < 2; ++j) {
        int n = j * 16 + l15;
        float bias = b0[n];
        #pragma unroll
        for (int v = 0; v < 8; ++v)
            act[v + half * 8][n] = gelu_gamma(acc0[j][v] + bias);  // C layout: row=v+half*8
    }
    __syncthreads();

    // ---- layer 1: [16,32] @ [32,32], 8 K-steps of 4 -------------------------
    v8f acc1[2] = {};
    #pragma unroll
    for (int kk = 0; kk < 8; ++kk) {
        int kb = kk * 4 + half * 2;
        v2f aa; aa.x = act[l15][kb]; aa.y = act[l15][kb + 1];
        #pragma unroll
        for (int j = 0; j < 2; ++j) {
            int n = j * 16 + l15;
            v2f bb; bb.x = W1[kb * FEAT + n]; bb.y = W1[(kb + 1) * FEAT + n];
            acc1[j] = __builtin_amdgcn_wmma_f32_16x16x4_f32(false, aa, false, bb,
                                                            (short)0, acc1[j], false, false);
        }
    }
    __syncthreads();   // all reads of act done before overwrite
    #pragma unroll
    for (int j = 0; j < 2; ++j) {
        int n = j * 16 + l15;
        float bias = b1[n];
        #pragma unroll
        for (int v = 0; v < 8; ++v)
            act[v + half * 8][n] = gelu_gamma(acc1[j][v] + bias);
    }
    __syncthreads();

    // ---- layer 2: [16,32] @ [32,3], N padded 3->16 --------------------------
    v8f accp = {};
    #pragma unroll
    for (int kk = 0; kk < 8; ++kk) {
        int kb = kk * 4 + half * 2;
        v2f aa; aa.x = act[l15][kb]; aa.y = act[l15][kb + 1];
        v2f bb;
        bb.x = (l15 < 3) ? Wp[kb * 3 + l15] : 0.0f;
        bb.y = (l15 < 3) ? Wp[(kb + 1) * 3 + l15] : 0.0f;
        accp = __builtin_amdgcn_wmma_f32_16x16x4_f32(false, aa, false, bb,
                                                     (short)0, accp, false, false);
    }
    if (l15 < 3) {
        float bias = bp[l15];
        #pragma unroll
        for (int v = 0; v < 8; ++v)
            angb[v + half * 8][l15] = accp[v] + bias;
    }
    __syncthreads();

    // ---- R = Rz @ Ry @ Rx, one sample per lane (lanes 0..15) ----------------
    if (lane < 16) {
        float ax = angb[lane][0], ay = angb[lane][1], az = angb[lane][2];
        float cx = cosf(ax), sx = sinf(ax);
        float cy = cosf(ay), sy = sinf(ay);
        float cz = cosf(az), sz = sinf(az);
        float* R = Rout + (size_t)(m0 + lane) * 9;
        R[0] = cz * cy;  R[1] = cz * sy * sx - sz * cx;  R[2] = cz * sy * cx + sz * sx;
        R[3] = sz * cy;  R[4] = sz * sy * sx + cz * cx;  R[5] = sz * sy * cx - cz * sx;
        R[6] = -sy;      R[7] = cy * sx;                 R[8] = cy * cx;
    }
}

// ---------------------------------------------------------------------------
// Kernel B: retina rays + monitor projection + bilinear gather (HBM-bound).
// grid = (H_OUT, N), block = W_OUT threads; one output pixel (3 channels)/thread.
// ---------------------------------------------------------------------------
__device__ __forceinline__ float tap(const float* __restrict__ p, int xi, int yi) {
    // zero pad-ring + clamp semantics of the reference, done analytically
    if ((unsigned)xi < (unsigned)W_IN && (unsigned)yi < (unsigned)H_IN)
        return p[yi * W_IN + xi];
    return 0.0f;
}

__launch_bounds__(W_OUT)
__global__ void MonitorRetina_sample(const float* __restrict__ stim,
                                     const float* __restrict__ Rmat,
                                     float* __restrict__ out) {
    const int x = threadIdx.x;       // 0..255
    const int y = blockIdx.x;        // 0..143
    const int n = blockIdx.y;        // 0..255

    // isotropic output grid (short side 144 spans ~[-1,1])
    const float inv_s = 1.0f / (float)H_OUT;
    float gx = ((x + 0.5f) * 2.0f - (float)W_OUT) * inv_s;
    float gy = ((y + 0.5f) * 2.0f - (float)H_OUT) * inv_s;

    float r      = sqrtf(gx * gx + gy * gy);
    float theta  = r * DEG2RAD_FOV;
    float safe_r = fmaxf(r, 1e-6f);
    float st = sinf(theta), ct = cosf(theta);
    float inv_r = 1.0f / safe_r;
    float dx = st * gx * inv_r;
    float dy = st * gy * inv_r;
    float dz = ct;

    // rotate by per-sample R (block-uniform -> scalar loads)
    const float* R = Rmat + (size_t)n * 9;
    float rx = R[0] * dx + R[1] * dy + R[2] * dz;
    float ry = R[3] * dx + R[4] * dy + R[5] * dz;
    float rz = R[6] * dx + R[7] * dy + R[8] * dz;

    float z  = fmaxf(rz, 1e-6f);
    float u  = rx / z;
    float v  = ry / z;

    // isotropic -> unpadded input pixel coords (s_in = 270)
    float fx = (u * ((float)H_IN / (float)W_IN) + 1.0f) * ((float)W_IN * 0.5f) - 0.5f;
    float fy = (v + 1.0f) * ((float)H_IN * 0.5f) - 0.5f;

    float x0f = floorf(fx), y0f = floorf(fy);
    int   x0 = (int)x0f,    y0 = (int)y0f;
    float wx = fx - x0f,    wy = fy - y0f;
    float w00 = (1.0f - wx) * (1.0f - wy);
    float w01 = wx * (1.0f - wy);
    float w10 = (1.0f - wx) * wy;
    float w11 = wx * wy;

    #pragma unroll
    for (int c = 0; c < C_IN; ++c) {
        const float* img = stim + ((size_t)(n * C_IN + c)) * (H_IN * W_IN);
        float v00 = tap(img, x0,     y0);
        float v01 = tap(img, x0 + 1, y0);
        float v10 = tap(img, x0,     y0 + 1);
        float v11 = tap(img, x0 + 1, y0 + 1);
        float o = v00 * w00 + v01 * w01 + v10 * w10 + v11 * w11;
        size_t oidx = (((size_t)(n * C_IN + c)) * H_OUT + y) * W_OUT + x;
        __builtin_nontemporal_store(o, &out[oidx]);   // write-once output
    }
}

extern "C" void kernel_launch(void* const* d_in, const int* in_sizes, int n_in,
                              void* d_out, int out_size, void* d_ws, size_t ws_size,
                              hipStream_t stream) {
    const float* stim = (const float*)d_in[0];
    const float* eye  = (const float*)d_in[1];
    const float* W0   = (const float*)d_in[2];
    const float* b0   = (const float*)d_in[3];
    const float* W1   = (const float*)d_in[4];
    const float* b1   = (const float*)d_in[5];
    const float* Wp   = (const float*)d_in[6];
    const float* bp   = (const float*)d_in[7];
    // d_in[8] = height, d_in[9] = width (fixed 144/256 in setup; compile-time here)

    float* Rws = (float*)d_ws;                 // 256*9 floats of scratch
    float* out = (float*)d_out;

    MonitorRetina_mlp_rmat<<<dim3(N_SAMPLES / 16), dim3(32), 0, stream>>>(
        eye, W0, b0, W1, b1, Wp, bp, Rws);

    MonitorRetina_sample<<<dim3(H_OUT, N_SAMPLES), dim3(W_OUT), 0, stream>>>(
        stim, Rws, out);
}